// SimpleLeakySNN_50130858279581
// MI455X (gfx1250) — compile-verified
//
#include <hip/hip_runtime.h>

typedef __attribute__((ext_vector_type(2))) float v2f;
typedef __attribute__((ext_vector_type(8))) float v8f;

#define T_STEPS 100
#define BATCH   4096
#define NI      128
#define NH      512
#define NO      16
#define BM      16            // batch rows per workgroup
#define W1T_LD  (NH + 8)      // 520 floats: padded to avoid LDS bank conflicts
#define SPK_LD  (NH + 8)      // 520

__device__ __forceinline__ v8f splat8(float s) {
    v8f v = {s, s, s, s, s, s, s, s};
    return v;
}

__global__ __launch_bounds__(256)
void snn_leaky_kernel(const float* __restrict__ x,
                      const float* __restrict__ W1,
                      const float* __restrict__ b1,
                      const float* __restrict__ W2,
                      const float* __restrict__ b2,
                      float* __restrict__ out)
{
    // LDS: W1 transposed [k=128][n=512] (padded), spike buffer [16][512] (padded),
    // cur2 reduction buffer [16][16].  Total ~300.5 KB of the 320 KB WGP LDS.
    __shared__ float s_w1t[NI * W1T_LD];
    __shared__ float s_spk1[BM * SPK_LD];
    __shared__ float s_cur2[BM * NO];

    const int tid  = threadIdx.x;
    const int wave = tid >> 5;
    const int lane = tid & 31;
    const int nl   = lane & 15;   // M (A) / N (B,C,D) index within tile
    const int hi   = lane >> 4;   // half-wave selector
    const int koff = hi * 2;      // K base for this half-wave in A/B fragments
    const int b0   = blockIdx.x * BM;
    const int wcol = wave * 64;   // this wave's hidden-column base (GEMM1)

    // ---- one-time: transpose W1 [NH,NI] into LDS as W1T[k][n] ----
    for (int idx = tid; idx < NH * NI; idx += 256) {
        int n = idx >> 7;          // / NI
        int k = idx & (NI - 1);
        s_w1t[k * W1T_LD + n] = W1[idx];
    }

    // ---- biases (per-lane scalars) ----
    float b1v[4];
    #pragma unroll
    for (int nb = 0; nb < 4; ++nb) b1v[nb] = b1[wcol + nb * 16 + nl];
    const float b2v = b2[nl];

    // ---- persistent W2 B-fragments: wave w owns K slice [w*64, w*64+64) ----
    v2f w2frag[16];
    {
        const float* w2p = W2 + (size_t)nl * NH + koff;   // W2 is [NO, NH] row-major
        #pragma unroll
        for (int i = 0; i < 16; ++i) {
            int kb = wave * 16 + i;
            w2frag[i] = *(const v2f*)(w2p + kb * 4);
        }
    }

    // ---- persistent membrane state ----
    v8f mem1[4];
    #pragma unroll
    for (int nb = 0; nb < 4; ++nb) mem1[nb] = splat8(0.0f);
    v8f mem2 = splat8(0.0f);

    float* out_spk = out;
    float* out_mem = out + (size_t)T_STEPS * BATCH * NO;

    __syncthreads();   // W1T ready

    for (int t = 0; t < T_STEPS; ++t) {
        const float* xt = x + ((size_t)t * BATCH + b0 + nl) * NI + koff;

        // Prefetch next timestep's x tile while we do this step's WMMA work.
        if (t + 1 < T_STEPS) {
            const char* pf = (const char*)(x + ((size_t)(t + 1) * BATCH + b0 + nl) * NI)
                             + hi * 256;
            __builtin_prefetch(pf, 0, 0);
            __builtin_prefetch(pf + 128, 0, 0);
        }

        // ================= layer 1: cur1 = x_t @ W1^T + b1 =================
        // f32 WMMA 16x16x4, K = 128 -> 32 k-blocks; each wave: 1 M-tile x 4 N-tiles
        v8f acc[4];
        #pragma unroll
        for (int nb = 0; nb < 4; ++nb) acc[nb] = splat8(b1v[nb]);   // seed with bias

        for (int kb = 0; kb < 32; ++kb) {
            v2f av = *(const v2f*)(xt + kb * 4);                    // A frag (global)
            const int krow = kb * 4 + koff;
            #pragma unroll
            for (int nb = 0; nb < 4; ++nb) {
                v2f bv;
                bv.x = s_w1t[(krow + 0) * W1T_LD + wcol + nb * 16 + nl];
                bv.y = s_w1t[(krow + 1) * W1T_LD + wcol + nb * 16 + nl];
                acc[nb] = __builtin_amdgcn_wmma_f32_16x16x4_f32(
                    false, av, false, bv, (short)0, acc[nb], false, false);
            }
        }

        __syncthreads();   // prior step's consumers of s_spk1 / s_cur2 are done

        s_cur2[tid] = 0.0f;   // 256 threads zero the 16x16 reduction buffer

        // membrane update layer 1 + spike store to LDS
        #pragma unroll
        for (int nb = 0; nb < 4; ++nb) {
            v8f m = mem1[nb];
            v8f c = acc[nb];
            #pragma unroll
            for (int r = 0; r < 8; ++r) {
                float mv    = m[r];
                float reset = (mv > 1.0f) ? 1.0f : 0.0f;     // from PREVIOUS mem
                float mn    = 0.95f * mv + c[r] - reset;
                float spk   = (mn > 1.0f) ? 1.0f : 0.0f;
                m[r] = mn;
                s_spk1[(r + 8 * hi) * SPK_LD + wcol + nb * 16 + nl] = spk;
            }
            mem1[nb] = m;
        }

        __syncthreads();   // spk1 + zeroed cur2 visible to all waves

        // ================= layer 2: cur2 = spk1 @ W2^T (K split over waves) ======
        v8f c2 = splat8(0.0f);
        #pragma unroll
        for (int i = 0; i < 16; ++i) {
            int kb = wave * 16 + i;
            v2f av = *(const v2f*)&s_spk1[nl * SPK_LD + kb * 4 + koff];
            c2 = __builtin_amdgcn_wmma_f32_16x16x4_f32(
                false, av, false, w2frag[i], (short)0, c2, false, false);
        }
        #pragma unroll
        for (int r = 0; r < 8; ++r) {
            atomicAdd(&s_cur2[(r + 8 * hi) * NO + nl], c2[r]);   // ds_add_f32
        }

        __syncthreads();   // cross-wave reduction complete

        // membrane update layer 2 + output stores (wave 0 owns mem2)
        if (wave == 0) {
            #pragma unroll
            for (int r = 0; r < 8; ++r) {
                int   row   = r + 8 * hi;
                float cur   = s_cur2[row * NO + nl] + b2v;
                float mv    = mem2[r];
                float reset = (mv > 1.0f) ? 1.0f : 0.0f;
                float mn    = 0.95f * mv + cur - reset;
                float spk   = (mn > 1.0f) ? 1.0f : 0.0f;
                mem2[r] = mn;
                size_t o = ((size_t)t * BATCH + b0 + row) * NO + nl;
                out_spk[o] = spk;
                out_mem[o] = mn;
            }
        }
    }
}

extern "C" void kernel_launch(void* const* d_in, const int* in_sizes, int n_in,
                              void* d_out, int out_size, void* d_ws, size_t ws_size,
                              hipStream_t stream) {
    const float* x  = (const float*)d_in[0];
    const float* W1 = (const float*)d_in[1];
    const float* b1 = (const float*)d_in[2];
    const float* W2 = (const float*)d_in[3];
    const float* b2 = (const float*)d_in[4];
    float* out = (float*)d_out;

    dim3 grid(BATCH / BM);   // 256 workgroups, each runs the full time loop
    dim3 block(256);         // 8 wave32 waves
    hipLaunchKernelGGL(snn_leaky_kernel, grid, block, 0, stream,
                       x, W1, b1, W2, b2, out);
}